// GraphSearchPolicy_23862838297041
// MI455X (gfx1250) — compile-verified
//
#include <hip/hip_runtime.h>

// ---------------------------------------------------------------------------
// Sizes (from reference)
// ---------------------------------------------------------------------------
#define BATCH 512
#define NRREL 400
#define NACT  256
#define ENTD  200
#define RELD  200
#define HISTD 400

typedef __attribute__((ext_vector_type(2))) float v2f;
typedef __attribute__((ext_vector_type(8))) float v8f;

#if defined(__HIP_DEVICE_COMPILE__) && __has_builtin(__builtin_amdgcn_wmma_f32_16x16x4_f32)
#define HAVE_WMMA_F32X4 1
#else
#define HAVE_WMMA_F32X4 0
#endif

__device__ __forceinline__ v8f wmma_f32x4(v2f a, v2f b, v8f c)
{
#if HAVE_WMMA_F32X4
    return __builtin_amdgcn_wmma_f32_16x16x4_f32(
        false, a, false, b, (short)0, c, false, false);
#else
    #pragma unroll
    for (int v = 0; v < 8; ++v) c[v] += a.x * b.x + a.y * b.y;
    return c;
#endif
}

// ---------------------------------------------------------------------------
// Wave-level WMMA GEMM:  C[M,N] = A[M,K] * op(B) (+bias) (+relu)
//   BT = true : B row-major [N,K]  (C = A * B^T)   -- torch weight layout
//   BT = false: B row-major [K,N]  (C = A * B)
// One wave computes a 16x32 strip of C (two 16x16 WMMA tiles sharing the A
// fragment), K-loop in steps of 4 with V_WMMA_F32_16X16X4_F32.
// M, K are multiples of 16/4 here.  Ragged N handled by ADDRESS CLAMPING on
// B/bias loads (an OOB B column only feeds an OOB C column, which is never
// stored), so all loads in the hot loop are unconditional -> no EXEC masking,
// and WMMA runs with EXEC all-ones as the ISA requires.
//
// VGPR layouts per CDNA5 ISA 7.12.2 (32-bit, wave32):
//   A 16x4 : lane L -> M = L%16,  vgpr v -> K = 2*(L/16) + v
//   B 4x16 : lane L -> N = L%16,  vgpr v -> K = 2*(L/16) + v
//   C 16x16: lane L -> N = L%16,  vgpr v -> M = v + 8*(L/16)
// ---------------------------------------------------------------------------
template <bool BT, bool RELU, bool BIAS>
__global__ void gemm_wmma_f32(const float* __restrict__ A,
                              const float* __restrict__ B,
                              const float* __restrict__ bias,
                              float* __restrict__ C,
                              int M, int N, int K,
                              int lda, int ldb, int ldc, int tilesN)
{
    const int wave = (blockIdx.x * blockDim.x + threadIdx.x) >> 5;
    const int lane = threadIdx.x & 31;
    const int tM = wave / tilesN;
    const int tN = wave % tilesN;          // one tile = 32 columns
    const int m0 = tM * 16;
    const int n0 = tN * 32;
    if (m0 >= M) return;                   // uniform per-wave exit

    const int half = lane >> 4;            // 0..1
    const int lm   = lane & 15;            // 0..15
    const int nn0  = n0 + lm;
    const int nn1  = n0 + 16 + lm;
    const int nc0  = (nn0 < N) ? nn0 : (N - 1);   // clamped (value never stored if OOB)
    const int nc1  = (nn1 < N) ? nn1 : (N - 1);

    const float* ap = A + (long)(m0 + lm) * lda + 2 * half;
    const float* bp0;
    const float* bp1;
    if (BT) {
        bp0 = B + (long)nc0 * ldb + 2 * half;
        bp1 = B + (long)nc1 * ldb + 2 * half;
    } else {
        bp0 = B + (long)(2 * half) * ldb + nc0;
        bp1 = B + (long)(2 * half) * ldb + nc1;
    }

    v8f acc0 = {};
    v8f acc1 = {};
    for (int k0 = 0; k0 < K; k0 += 4) {
        v2f a, b0, b1;
        a.x = ap[0]; a.y = ap[1]; ap += 4;
        if (BT) {
            b0.x = bp0[0]; b0.y = bp0[1]; bp0 += 4;
            b1.x = bp1[0]; b1.y = bp1[1]; bp1 += 4;
        } else {
            b0.x = bp0[0]; b0.y = bp0[ldb];
            b1.x = bp1[0]; b1.y = bp1[ldb];
            bp0 += 4 * (long)ldb;
            bp1 += 4 * (long)ldb;
        }
        acc0 = wmma_f32x4(a, b0, acc0);    // two independent accumulate chains
        acc1 = wmma_f32x4(a, b1, acc1);
    }

    float bv0 = 0.0f, bv1 = 0.0f;
    if (BIAS) { bv0 = bias[nc0]; bv1 = bias[nc1]; }

    if (nn0 < N) {
        #pragma unroll
        for (int v = 0; v < 8; ++v) {
            float x = acc0[v] + bv0;
            if (RELU) x = fmaxf(x, 0.0f);
            C[(long)(m0 + v + 8 * half) * ldc + nn0] = x;
        }
    }
    if (nn1 < N) {
        #pragma unroll
        for (int v = 0; v < 8; ++v) {
            float x = acc1[v] + bv1;
            if (RELU) x = fmaxf(x, 0.0f);
            C[(long)(m0 + v + 8 * half) * ldc + nn1] = x;
        }
    }
}

// ---------------------------------------------------------------------------
// Gather / concat kernels
// ---------------------------------------------------------------------------
__global__ void build_xin_k(const float* __restrict__ H,
                            const float* __restrict__ ent,
                            const float* __restrict__ rel,
                            const int* __restrict__ e,
                            const int* __restrict__ q,
                            float* __restrict__ Xin)   // [B, 800] = [E|H|Q]
{
    const int b = blockIdx.x;
    const float* E = ent + (long)e[b] * ENTD;
    const float* Q = rel + (long)q[b] * RELD;
    float* o = Xin + (long)b * 800;
    for (int i = threadIdx.x; i < ENTD;  i += blockDim.x) o[i]                = E[i];
    for (int i = threadIdx.x; i < HISTD; i += blockDim.x) o[ENTD + i]         = H[(long)b * HISTD + i];
    for (int i = threadIdx.x; i < RELD;  i += blockDim.x) o[ENTD + HISTD + i] = Q[i];
}

__global__ void gather_pred_k(const float* __restrict__ ent,
                              const int* __restrict__ pid,
                              float* __restrict__ pe)  // [B, 200]
{
    const int b = blockIdx.x;
    const float* src = ent + (long)pid[b] * ENTD;
    for (int i = threadIdx.x; i < ENTD; i += blockDim.x)
        pe[(long)b * ENTD + i] = src[i];
}

__global__ void rel_sum_k(const float* __restrict__ rel, float* __restrict__ rs)
{
    const int d = blockIdx.x * blockDim.x + threadIdx.x;
    if (d < RELD) {
        float s = 0.0f;
        for (int r = 0; r < NRREL; ++r) s += rel[(long)r * RELD + d];
        rs[d] = s;
    }
}

__global__ void build_alpha_in_k(const float* __restrict__ rs,
                                 const float* __restrict__ pe,
                                 float* __restrict__ ai)   // [B, 400]
{
    const int b = blockIdx.x;
    for (int d = threadIdx.x; d < RELD + ENTD; d += blockDim.x)
        ai[(long)b * 400 + d] =
            (d < RELD) ? rs[d] : (float)NRREL * pe[(long)b * ENTD + (d - RELD)];
}

// ---------------------------------------------------------------------------
// Row softmax (block per row). In-place safe.
// ---------------------------------------------------------------------------
__global__ void softmax_rows_k(const float* __restrict__ src,
                               float* __restrict__ dst, int N, int stride)
{
    const int b = blockIdx.x;
    const float* x = src + (long)b * stride;
    float* y = dst + (long)b * stride;
    __shared__ float red[256];
    const int t = threadIdx.x;

    float mx = -3.0e38f;
    for (int i = t; i < N; i += 256) mx = fmaxf(mx, x[i]);
    red[t] = mx; __syncthreads();
    for (int s = 128; s > 0; s >>= 1) { if (t < s) red[t] = fmaxf(red[t], red[t + s]); __syncthreads(); }
    mx = red[0]; __syncthreads();

    float sum = 0.0f;
    for (int i = t; i < N; i += 256) sum += expf(x[i] - mx);
    red[t] = sum; __syncthreads();
    for (int s = 128; s > 0; s >>= 1) { if (t < s) red[t] += red[t + s]; __syncthreads(); }
    const float inv = 1.0f / red[0];

    for (int i = t; i < N; i += 256) y[i] = expf(x[i] - mx) * inv;
}

// ---------------------------------------------------------------------------
// scores[b,a] = dot(rel[r_space[b,a]], X2[b,0:200]) + dot(ent[e_space[b,a]], X2[b,200:400])
// one wave per (b,a); X2 lives in Tin with row stride 600 (cols 0..400)
// ---------------------------------------------------------------------------
__global__ void scores_k(const float* __restrict__ rel,
                         const float* __restrict__ ent,
                         const int* __restrict__ r_space,
                         const int* __restrict__ e_space,
                         const float* __restrict__ X2,   // stride 600
                         float* __restrict__ scores)
{
    const int gw   = (blockIdx.x * blockDim.x + threadIdx.x) >> 5;
    const int lane = threadIdx.x & 31;
    const int b = gw >> 8;          // NACT == 256
    const int a = gw & (NACT - 1);

    const float* R = rel + (long)r_space[(long)b * NACT + a] * RELD;
    const float* E = ent + (long)e_space[(long)b * NACT + a] * ENTD;
    const float* x = X2 + (long)b * 600;

    float s = 0.0f;
    for (int d = lane; d < RELD; d += 32)
        s += R[d] * x[d] + E[d] * x[RELD + d];
    #pragma unroll
    for (int off = 16; off > 0; off >>= 1) s += __shfl_xor(s, off, 32);
    if (lane == 0) scores[(long)b * NACT + a] = s;
}

// ---------------------------------------------------------------------------
// Masked softmax over 256 actions + entropy; block of 256 per batch row
// ---------------------------------------------------------------------------
__global__ void action_k(const float* __restrict__ scores,
                         const float* __restrict__ mask,
                         float* __restrict__ dist,
                         float* __restrict__ entropy)
{
    const int b = blockIdx.x, a = threadIdx.x;
    __shared__ float red[256];

    float sc = scores[(long)b * NACT + a] - (1.0f - mask[(long)b * NACT + a]) * 1.0e31f;

    red[a] = sc; __syncthreads();
    for (int s = 128; s > 0; s >>= 1) { if (a < s) red[a] = fmaxf(red[a], red[a + s]); __syncthreads(); }
    const float mx = red[0]; __syncthreads();

    const float ex = expf(sc - mx);
    red[a] = ex; __syncthreads();
    for (int s = 128; s > 0; s >>= 1) { if (a < s) red[a] += red[a + s]; __syncthreads(); }
    const float p = ex / red[0]; __syncthreads();

    dist[(long)b * NACT + a] = p;

    red[a] = -p * logf(p + 1.0e-20f); __syncthreads();
    for (int s = 128; s > 0; s >>= 1) { if (a < s) red[a] += red[a + s]; __syncthreads(); }
    if (a == 0) entropy[b] = red[0];
}

// ---------------------------------------------------------------------------
// Host launcher
// ---------------------------------------------------------------------------
static inline int gemm_blocks(int M, int N)
{
    const int tiles = ((M + 15) / 16) * ((N + 31) / 32);  // 16x32 strip per wave
    return (tiles + 3) / 4;                               // 4 waves per block
}
static inline int tiles_n(int N) { return (N + 31) / 32; }

extern "C" void kernel_launch(void* const* d_in, const int* in_sizes, int n_in,
                              void* d_out, int out_size, void* d_ws, size_t ws_size,
                              hipStream_t stream)
{
    (void)in_sizes; (void)n_in; (void)out_size; (void)ws_size;

    const float* H     = (const float*)d_in[0];
    const float* amask = (const float*)d_in[1];
    const float* ent   = (const float*)d_in[2];
    const float* rel   = (const float*)d_in[3];
    const float* W1 = (const float*)d_in[4];  const float* b1 = (const float*)d_in[5];
    const float* W2 = (const float*)d_in[6];  const float* b2 = (const float*)d_in[7];
    const float* W3 = (const float*)d_in[8];  const float* b3 = (const float*)d_in[9];
    const float* W4 = (const float*)d_in[10]; const float* b4 = (const float*)d_in[11];
    const float* W5 = (const float*)d_in[12]; const float* b5 = (const float*)d_in[13];
    const float* W6 = (const float*)d_in[14]; const float* b6 = (const float*)d_in[15];
    const float* Watt = (const float*)d_in[16]; const float* batt = (const float*)d_in[17];
    const int* e       = (const int*)d_in[18];
    const int* q       = (const int*)d_in[19];
    const int* pred_id = (const int*)d_in[20];
    const int* r_space = (const int*)d_in[21];
    const int* e_space = (const int*)d_in[22];

    float* out      = (float*)d_out;
    float* out_dist = out;                          // [512,256]
    float* out_ent  = out + BATCH * NACT;           // [512]
    float* out_att  = out + BATCH * NACT + BATCH;   // [512,400]

    float* ws = (float*)d_ws;
    float* Xin  = ws;                 // [512,800]
    float* X    = Xin  + 512 * 800;   // [512,400]
    float* Tin  = X    + 512 * 400;   // [512,600]  cols 0..400 = X2, 400..600 = V_A
    float* PE   = Tin  + 512 * 600;   // [512,200]  pred_emb
    float* PEI  = PE   + 512 * 200;   // [512,200]  pred_ent_info
    float* RK   = PEI  + 512 * 200;   // [400,200]  relation_K
    float* P    = RK   + 400 * 200;   // [512,400]  local_att -> softmax (in place)
    float* VAin = P    + 512 * 400;   // [512,400]  cols 0..200 = alpha, 200..400 = beta
    float* AIn  = VAin + 512 * 400;   // [512,400]  alpha_in
    float* T    = AIn  + 512 * 400;   // [512,200]
    float* SC   = T    + 512 * 200;   // [512,256]  scores
    float* RS   = SC   + 512 * 256;   // [200]      rel_sum

    const dim3 B128(128), B256(256);

    // 1) Xin = [E | H | Q]
    build_xin_k<<<BATCH, B256, 0, stream>>>(H, ent, rel, e, q, Xin);
    // 2) X = relu(Xin @ W1^T + b1)              [512,400]
    gemm_wmma_f32<true, true, true><<<gemm_blocks(512, 400), B128, 0, stream>>>(
        Xin, W1, b1, X, 512, 400, 800, 800, 800, 400, tiles_n(400));
    // 3) X2 = X @ W2^T + b2  -> Tin[:, 0:400]
    gemm_wmma_f32<true, false, true><<<gemm_blocks(512, 400), B128, 0, stream>>>(
        X, W2, b2, Tin, 512, 400, 400, 400, 400, 600, tiles_n(400));
    // 4) pred_emb gather
    gather_pred_k<<<BATCH, B256, 0, stream>>>(ent, pred_id, PE);
    // 5) pred_ent_info = pred_emb @ W3^T + b3   [512,200]
    gemm_wmma_f32<true, false, true><<<gemm_blocks(512, 200), B128, 0, stream>>>(
        PE, W3, b3, PEI, 512, 200, 200, 200, 200, 200, tiles_n(200));
    // 6) relation_K = relation_emb @ W4^T + b4  [400,200]
    gemm_wmma_f32<true, false, true><<<gemm_blocks(400, 200), B128, 0, stream>>>(
        rel, W4, b4, RK, 400, 200, 200, 200, 200, 200, tiles_n(200));
    // 7) local_att = pred_ent_info @ relation_K^T   [512,400]
    gemm_wmma_f32<true, false, false><<<gemm_blocks(512, 400), B128, 0, stream>>>(
        PEI, RK, nullptr, P, 512, 400, 200, 200, 200, 400, tiles_n(400));
    // 8) softmax over NR (in place)
    softmax_rows_k<<<BATCH, B256, 0, stream>>>(P, P, 400, 400);
    // 9) rel_sum
    rel_sum_k<<<1, B256, 0, stream>>>(rel, RS);
    // 10) alpha_in = [rel_sum | NR * pred_emb]
    build_alpha_in_k<<<BATCH, B256, 0, stream>>>(RS, PE, AIn);
    // 11) beta = P @ relation_emb  -> VAin[:, 200:400]    (B row-major [K=400, N=200])
    gemm_wmma_f32<false, false, false><<<gemm_blocks(512, 200), B128, 0, stream>>>(
        P, rel, nullptr, VAin + 200, 512, 200, 400, 400, 200, 400, tiles_n(200));
    // 12) alpha = alpha_in @ W5^T + b5  -> VAin[:, 0:200]
    gemm_wmma_f32<true, false, true><<<gemm_blocks(512, 200), B128, 0, stream>>>(
        AIn, W5, b5, VAin, 512, 200, 400, 400, 400, 400, tiles_n(200));
    // 13) V_A = VAin @ W6^T + b6  -> Tin[:, 400:600]
    gemm_wmma_f32<true, false, true><<<gemm_blocks(512, 200), B128, 0, stream>>>(
        VAin, W6, b6, Tin + 400, 512, 200, 400, 400, 400, 600, tiles_n(200));
    // 14) T = Tin @ Watt^T + batt   [512,200]
    gemm_wmma_f32<true, false, true><<<gemm_blocks(512, 200), B128, 0, stream>>>(
        Tin, Watt, batt, T, 512, 200, 600, 600, 600, 200, tiles_n(200));
    // 15) att logits = T @ relation_emb^T  -> out_att [512,400]
    gemm_wmma_f32<true, false, false><<<gemm_blocks(512, 400), B128, 0, stream>>>(
        T, rel, nullptr, out_att, 512, 400, 200, 200, 200, 400, tiles_n(400));
    // 16) relation_att = softmax(logits) in place
    softmax_rows_k<<<BATCH, B256, 0, stream>>>(out_att, out_att, 400, 400);
    // 17) gather-dot scores (one wave per (b,a))
    scores_k<<<(BATCH * NACT) / 8, B256, 0, stream>>>(rel, ent, r_space, e_space, Tin, SC);
    // 18) masked softmax + entropy
    action_k<<<BATCH, B256, 0, stream>>>(SC, amask, out_dist, out_ent);
}